// LearnedScalarAttentionPure_15040975470961
// MI455X (gfx1250) — compile-verified
//
#include <hip/hip_runtime.h>

#define HIDDEN 128

typedef __attribute__((ext_vector_type(2))) float v2f;
typedef __attribute__((ext_vector_type(8))) float v8f;

// Monotone order-preserving float<->uint mapping for integer atomicMax on floats.
__device__ __forceinline__ unsigned f32_enc(float f) {
  unsigned u = __float_as_uint(f);
  return (u & 0x80000000u) ? ~u : (u | 0x80000000u);
}
__device__ __forceinline__ float f32_dec(unsigned u) {
  return (u & 0x80000000u) ? __uint_as_float(u & 0x7FFFFFFFu) : __uint_as_float(~u);
}

// Native non-returning f32 atomic add (GLOBAL_ATOMIC_ADD_F32), device scope so the
// RMW is coherent across WGPs; tracked by STOREcnt only (fire-and-forget).
__device__ __forceinline__ void atomic_add_f32_dev(float* p, float v) {
  asm volatile("global_atomic_add_f32 %0, %1, off scope:SCOPE_DEV"
               :: "v"(p), "v"(v) : "memory");
}
__device__ __forceinline__ void atomic_add_f32_dev_off(float* p, float v, int /*imm via template*/);

// Pass 0: zero output / denominators, reset segment-max sentinels (0 == "empty" -> 0.0).
__global__ void k_init(float* __restrict__ out, float* __restrict__ denom,
                       unsigned* __restrict__ smax, int nHE) {
  long i = (long)blockIdx.x * blockDim.x + threadIdx.x;
  long total = (long)nHE * HIDDEN;
  if (i < total) out[i] = 0.0f;
  if (i < nHE) { denom[i] = 0.0f; smax[i] = 0u; }
}

// Pass 1: node_scores[n] = dot(feats[n,:], W) via V_WMMA_F32_16X16X4_F32.
// One wave -> 16 nodes. A tile: 16x4 f32 per ISA layout (lane L: M=L%16, Kpair=2*(L/16)).
// B tile: W chunk replicated across all 16 N-columns, so every column of D equals the scores.
__global__ void k_node_scores(const float* __restrict__ feats,
                              const float* __restrict__ W,
                              float* __restrict__ scores, int nNodes) {
  const int lane = threadIdx.x & 31;
  const int wave = threadIdx.x >> 5;
  const int base = (blockIdx.x * 4 + wave) * 16;     // 4 waves / 128-thread block
  const int m    = lane & 15;
  const int hi   = lane >> 4;                        // 0: K=0,1  1: K=2,3 within each K4 chunk
  int row = base + m;
  if (row > nNodes - 1) row = nNodes - 1;            // clamp: keep EXEC all-1s for WMMA
  const float* fp = feats + (long)row * HIDDEN;

  v8f c = {};
  #pragma unroll
  for (int kb = 0; kb < HIDDEN; kb += 4) {
    v2f a, b;
    a.x = fp[kb + 2 * hi];
    a.y = fp[kb + 2 * hi + 1];
    b.x = W[kb + 2 * hi];        // replicated over N -> layout-robust
    b.y = W[kb + 2 * hi + 1];
    c = __builtin_amdgcn_wmma_f32_16x16x4_f32(false, a, false, b, (short)0, c,
                                              false, false);
  }
  // C/D layout: VGPR r, lanes 0-15 -> (M=r, N=lane); lanes 16-31 -> (M=8+r, N=lane-16).
  // Column N=0 lives in lane 0 (M=0..7) and lane 16 (M=8..15).
  if (m == 0) {
    int r0 = base + hi * 8;
    #pragma unroll
    for (int i = 0; i < 8; i++)
      if (r0 + i < nNodes) scores[r0 + i] = c[i];
  }
}

// Pass 2: segment max via integer atomicMax on encoded floats (native global_atomic_max_u32).
__global__ void k_segmax(const int* __restrict__ hidx, const float* __restrict__ scores,
                         unsigned* __restrict__ smax, long E) {
  long e = (long)blockIdx.x * blockDim.x + threadIdx.x;
  if (e >= E) return;
  int node = hidx[e];
  int he   = hidx[E + e];
  unsigned enc = f32_enc(scores[node]);
  (void)__hip_atomic_fetch_max(&smax[he], enc, __ATOMIC_RELAXED, __HIP_MEMORY_SCOPE_AGENT);
}

// Pass 3: denom[he] += exp(score - segmax)
__global__ void k_denom(const int* __restrict__ hidx, const float* __restrict__ scores,
                        const unsigned* __restrict__ smax, float* __restrict__ denom, long E) {
  long e = (long)blockIdx.x * blockDim.x + threadIdx.x;
  if (e >= E) return;
  int node = hidx[e];
  int he   = hidx[E + e];
  unsigned u = smax[he];
  float mx = (u == 0u) ? 0.0f : f32_dec(u);
  float w = expf(scores[node] - mx);
  atomic_add_f32_dev(&denom[he], w);
}

// Pass 4: out[he,:] += (w/denom) * feats[node,:].
// 32 lanes per edge, float4 per lane: per-edge metadata is wave-uniform, feats read as
// coalesced b128, and each lane fires 4 consecutive-offset native f32 atomics into the
// L2-resident output (immediate offsets share one VGPR address pair).
__global__ void k_scatter(const int* __restrict__ hidx, const float4* __restrict__ feats4,
                          const float* __restrict__ scores, const unsigned* __restrict__ smax,
                          const float* __restrict__ denom, float* __restrict__ out, long E) {
  long gid = (long)blockIdx.x * blockDim.x + threadIdx.x;
  long e = gid >> 5;                 // 32 lanes per edge -> e uniform within a wave
  int  q = (int)(gid & 31);          // float4 slot: features [4q, 4q+3]
  if (e >= E) return;
  int node = hidx[e];
  int he   = hidx[E + e];
  unsigned u = smax[he];
  float mx = (u == 0u) ? 0.0f : f32_dec(u);
  float w  = expf(scores[node] - mx);
  float d  = fmaxf(denom[he], 1e-20f);
  float wn = w / d;
  float4 v = feats4[(long)node * (HIDDEN / 4) + q];
  float* p = out + (long)he * HIDDEN + q * 4;
  asm volatile("global_atomic_add_f32 %0, %1, off scope:SCOPE_DEV"
               :: "v"(p), "v"(wn * v.x) : "memory");
  asm volatile("global_atomic_add_f32 %0, %1, off offset:4 scope:SCOPE_DEV"
               :: "v"(p), "v"(wn * v.y) : "memory");
  asm volatile("global_atomic_add_f32 %0, %1, off offset:8 scope:SCOPE_DEV"
               :: "v"(p), "v"(wn * v.z) : "memory");
  asm volatile("global_atomic_add_f32 %0, %1, off offset:12 scope:SCOPE_DEV"
               :: "v"(p), "v"(wn * v.w) : "memory");
}

extern "C" void kernel_launch(void* const* d_in, const int* in_sizes, int n_in,
                              void* d_out, int out_size, void* d_ws, size_t ws_size,
                              hipStream_t stream) {
  const float* feats = (const float*)d_in[0];
  const int*   hidx  = (const int*)d_in[1];   // [2,E] flat: row0 = node idx, row1 = he idx
  const float* W     = (const float*)d_in[3]; // [1,128]
  float* out = (float*)d_out;

  const int  nNodes = in_sizes[0] / HIDDEN;
  const long E      = (long)in_sizes[1] / 2;
  const int  nHE    = out_size / HIDDEN;

  // Workspace carve-out (256B aligned): scores[nNodes], smax[nHE], denom[nHE]
  char* ws = (char*)d_ws;
  size_t o0 = 0;
  size_t o1 = (o0 + (size_t)nNodes * 4 + 255) & ~(size_t)255;
  size_t o2 = (o1 + (size_t)nHE * 4 + 255) & ~(size_t)255;
  float*    scores = (float*)(ws + o0);
  unsigned* smax   = (unsigned*)(ws + o1);
  float*    denom  = (float*)(ws + o2);

  { // init
    long total = (long)nHE * HIDDEN;
    int blocks = (int)((total + 255) / 256);
    k_init<<<blocks, 256, 0, stream>>>(out, denom, smax, nHE);
  }
  { // WMMA GEMV: 64 nodes per 128-thread block (4 waves x 16 nodes)
    int blocks = (nNodes + 63) / 64;
    k_node_scores<<<blocks, 128, 0, stream>>>(feats, W, scores, nNodes);
  }
  { // segment max
    int blocks = (int)((E + 255) / 256);
    k_segmax<<<blocks, 256, 0, stream>>>(hidx, scores, smax, E);
  }
  { // denominators
    int blocks = (int)((E + 255) / 256);
    k_denom<<<blocks, 256, 0, stream>>>(hidx, scores, smax, denom, E);
  }
  { // weighted scatter: 32 lanes x float4 per edge
    long total = E * 32;
    int blocks = (int)((total + 255) / 256);
    k_scatter<<<blocks, 256, 0, stream>>>(hidx, (const float4*)feats, scores, smax,
                                          denom, out, E);
  }
}